// ARL_46694884442612
// MI455X (gfx1250) — compile-verified
//
#include <hip/hip_runtime.h>
#include <cstddef>

#define SEQ   500
#define EMB   300
#define NASP  5
#define H1    10
#define AH    50          // NASP*H1 (logical N)
#define AHP   64          // N padded to 4 WMMA tiles
#define SEQP  512
#define CTXW  30          // CTX*H1

// Weights in LDS stored as interleaved e-pairs: Wp[k2][n][2], k2 = e/2.
// Row stride 160 floats (128 used + 32 skew) so the lanes16-31 half
// (next k2 row, +160 % 64banks = 32) uses banks 32..63 -> conflict-free b64.
#define K2ROWS   (EMB / 2)         // 150
#define K2STRIDE 160
#define WP_SZ    (K2ROWS * K2STRIDE)   // 24000 floats

#define P_ROWS  (SEQ + 2)          // zero guard rows for +-1 window
#define P_SZ    (P_ROWS * AH)      // 25100
#define S_SZ    (NASP * SEQP)      // 2560
#define E_SZ    160                // aspect embeddings (150 used)
#define RED_SZ  16
#define RP_SZ   256                // pooled-rep partials (5 chunks x 50)
#define LDS_FLOATS (WP_SZ + P_SZ + S_SZ + E_SZ + RED_SZ + RP_SZ)

typedef float v2f __attribute__((ext_vector_type(2)));
typedef float v8f __attribute__((ext_vector_type(8)));

__global__ __launch_bounds__(256)
void arl_fused(const float* __restrict__ doc,     // [B,500,300]
               const float* __restrict__ projW,   // [5,300,10]
               const float* __restrict__ aspE,    // [5,30]
               float* __restrict__ outAttn,       // [B,5,500]
               float* __restrict__ outRep) {      // [B,5,10]
  extern __shared__ float smem[];
  float* Wp    = smem;              // [150][160] interleaved e-pairs
  float* P     = Wp + WP_SZ;        // [502][50]  proj with guard rows
  float* S     = P  + P_SZ;         // [5][512]   scores / attn
  float* E     = S  + S_SZ;         // [5][30]
  float* red   = E  + E_SZ;         // inv-sum per aspect
  float* rpart = red + RED_SZ;      // rep partials

  const int b    = blockIdx.x;
  const int tid  = threadIdx.x;
  const int wave = tid >> 5;
  const int lane = tid & 31;
  const int lh   = lane & 15;
  const bool hi  = lane >= 16;

  // ---- Phase 0: stage interleaved weights + embeddings, zero guard rows ----
  for (int i = tid; i < WP_SZ; i += 256) {
    int k2 = i / K2STRIDE, r = i % K2STRIDE;
    float v = 0.f;
    if (r < 2 * AHP) {
      int n = r >> 1, e = k2 * 2 + (r & 1);
      if (n < AH) v = projW[(n / H1) * (EMB * H1) + e * H1 + (n % H1)];
    }
    Wp[i] = v;
  }
  for (int i = tid; i < E_SZ; i += 256) E[i] = (i < NASP * CTXW) ? aspE[i] : 0.f;
  if (tid < AH) { P[tid] = 0.f; P[(SEQ + 1) * AH + tid] = 0.f; }
  __syncthreads();

  // ---- Phase 1: proj = doc x W via V_WMMA_F32_16X16X4_F32 ----
  const float* docB = doc + (size_t)b * (SEQ * EMB);
  for (int mt = wave; mt < 32; mt += 8) {          // 32 M-tiles of 16 seq rows
    const int s0 = mt * 16;
    const int sA = s0 + lh;                        // A row held by this lane
    // Clamp OOB rows to a valid row: garbage only reaches discarded D rows
    // (each D row depends solely on its own A row), so no masking needed.
    const int sClamped = (sA < SEQ) ? sA : (SEQ - 1);
    const float* arow = docB + (size_t)sClamped * EMB + (hi ? 2 : 0);

    v8f acc0 = {}, acc1 = {}, acc2 = {}, acc3 = {};
    for (int k0 = 0; k0 < EMB; k0 += 4) {
      // A 16x4 f32 frag: v0 = K=0 (lanes0-15)/K=2 (lanes16-31), v1 = K=1/K=3
      v2f af = *(const v2f*)(arow + k0);           // one global_load_b64
      // B 4x16 frags: (e, e+1) pair is contiguous -> single ds_load_b64 each
      const float* bp = Wp + ((k0 >> 1) + (hi ? 1 : 0)) * K2STRIDE + 2 * lh;
      v2f bf0 = *(const v2f*)(bp);
      v2f bf1 = *(const v2f*)(bp + 32);
      v2f bf2 = *(const v2f*)(bp + 64);
      v2f bf3 = *(const v2f*)(bp + 96);
      acc0 = __builtin_amdgcn_wmma_f32_16x16x4_f32(false, af, false, bf0, (short)0, acc0, false, false);
      acc1 = __builtin_amdgcn_wmma_f32_16x16x4_f32(false, af, false, bf1, (short)0, acc1, false, false);
      acc2 = __builtin_amdgcn_wmma_f32_16x16x4_f32(false, af, false, bf2, (short)0, acc2, false, false);
      acc3 = __builtin_amdgcn_wmma_f32_16x16x4_f32(false, af, false, bf3, (short)0, acc3, false, false);
    }
    // D layout: VGPR r -> M = r (lanes0-15) / r+8 (lanes16-31), N = lane&15
    const int sBase = s0 + (hi ? 8 : 0);
#pragma unroll
    for (int r = 0; r < 8; ++r) {
      int s = sBase + r;
      if (s < SEQ) {
        float* row = P + (s + 1) * AH;
        row[lh]      = acc0[r];
        row[16 + lh] = acc1[r];
        row[32 + lh] = acc2[r];
        if (lh < 2) row[48 + lh] = acc3[r];
      }
    }
  }
  __syncthreads();

  // ---- Phase 2: window scores  S[a][s] = sum_c sum_h P[s-1+c][a*10+h]*E[a][c*10+h]
  for (int idx = tid; idx < NASP * SEQ; idx += 256) {
    int a = idx / SEQ, s = idx % SEQ;
    const float* ea = E + a * CTXW;
    float sc = 0.f;
#pragma unroll
    for (int c = 0; c < 3; ++c) {
      const float* pr = P + (s + c) * AH + a * H1;   // guard rows absorb s-1, s+1
      const float* ec = ea + c * H1;
#pragma unroll
      for (int h = 0; h < H1; ++h) sc += pr[h] * ec[h];
    }
    S[a * SEQP + s] = sc;
  }
  __syncthreads();

  // ---- Phase 3: softmax over seq, one wave per aspect ----
  if (wave < NASP) {
    float m = -3.402823466e+38f;
    for (int s = lane; s < SEQ; s += 32) m = fmaxf(m, S[wave * SEQP + s]);
    for (int off = 16; off; off >>= 1) m = fmaxf(m, __shfl_xor(m, off, 32));
    float sum = 0.f;
    for (int s = lane; s < SEQ; s += 32) {
      float ev = __expf(S[wave * SEQP + s] - m);
      S[wave * SEQP + s] = ev;
      sum += ev;
    }
    for (int off = 16; off; off >>= 1) sum += __shfl_xor(sum, off, 32);
    if (lane == 0) red[wave] = 1.0f / sum;
  }
  __syncthreads();

  // ---- Phase 4: normalize + write attention ----
  float* oA = outAttn + (size_t)b * (NASP * SEQ);
  for (int idx = tid; idx < NASP * SEQ; idx += 256) {
    int a = idx / SEQ, s = idx % SEQ;
    float v = S[a * SEQP + s] * red[a];
    S[a * SEQP + s] = v;
    oA[idx] = v;                                    // idx == a*500 + s
  }
  __syncthreads();

  // ---- Phase 5: pooled rep[a][h] = sum_s P[s][a*10+h] * attn[a][s] ----
  {
    int c = tid / AH;                // seq chunk 0..4 (threads >= 250 idle)
    int o = tid % AH;                // o == a*10 + h
    if (c < 5) {
      int a = o / H1;
      float r = 0.f;
      int sBeg = c * 100, sEnd = sBeg + 100;
      for (int s = sBeg; s < sEnd; ++s)
        r += P[(s + 1) * AH + o] * S[a * SEQP + s];
      rpart[c * AH + o] = r;
    }
  }
  __syncthreads();
  if (tid < AH) {
    float r = rpart[tid] + rpart[AH + tid] + rpart[2 * AH + tid] +
              rpart[3 * AH + tid] + rpart[4 * AH + tid];
    outRep[(size_t)b * AH + tid] = r;
  }
}

extern "C" void kernel_launch(void* const* d_in, const int* in_sizes, int n_in,
                              void* d_out, int out_size, void* d_ws, size_t ws_size,
                              hipStream_t stream) {
  const float* doc   = (const float*)d_in[0];
  const float* projW = (const float*)d_in[1];
  const float* aspE  = (const float*)d_in[2];
  float* out = (float*)d_out;

  const int nb = in_sizes[0] / (SEQ * EMB);           // 1024 docs
  float* outRep = out + (size_t)nb * (NASP * SEQ);    // rep follows attn

  const size_t ldsBytes = (size_t)LDS_FLOATS * sizeof(float);  // ~208 KB < 320 KB WGP LDS
  (void)hipFuncSetAttribute(reinterpret_cast<const void*>(arl_fused),
                            hipFuncAttributeMaxDynamicSharedMemorySize,
                            (int)ldsBytes);

  arl_fused<<<nb, 256, ldsBytes, stream>>>(doc, projW, aspE, out, outRep);
}